// SplineBinaryEncoding_75969381532163
// MI455X (gfx1250) — compile-verified
//
#include <hip/hip_runtime.h>
#include <hip/hip_bf16.h>

// ---------------------------------------------------------------------------
// SplineBinaryEncoding for MI455X (gfx1250)
//
// out[f,d] = sum over 6 scales x 2 coords of (1-a)*w[idx,d] + a*w[idx+1,d]
// L2-gather bound (~1.26 GB of 400B row gathers) + 52 MB streamed output.
// Not a GEMM (gathered rows are fragment-dependent) -> no WMMA; the CDNA5
// win is data movement: async global->LDS staging of the hot coarse tables
// (ASYNCcnt), saddr-form b128 gathers, NT output stores, high occupancy.
// ---------------------------------------------------------------------------

#define GLOBAL_AS __attribute__((address_space(1)))
#define LOCAL_AS  __attribute__((address_space(3)))

typedef int   v4i_t __attribute__((__vector_size__(4 * sizeof(int))));
typedef float v4f   __attribute__((ext_vector_type(4)));

namespace {
constexpr int F_TOT   = 131072;
constexpr int D_EMB   = 100;
constexpr int FPB     = 128;            // fragments per block
constexpr int BLOCK   = 256;            // threads per block (8 wave32)
constexpr int WPB     = BLOCK / 32;     // 8 waves -> 8 concurrent fragments
constexpr int JITERS  = FPB / WPB;      // 16 fragment iterations per wave

// Bin tables (compile-time; fully folded after unrolling)
__device__ constexpr int   kBW[6]    = {100, 200, 500, 1000, 2000, 5000};
__device__ constexpr int   kSHIFT[6] = {-1000, -500, -200, -100, -50, -20};
__device__ constexpr int   kCUM[6]   = {0, 2001, 3002, 3403, 3604, 3705};
__device__ constexpr float kINV[6]   = {1.0f / 100, 1.0f / 200, 1.0f / 500,
                                        1.0f / 1000, 1.0f / 2000, 1.0f / 5000};

// Scales 4 and 5 live in LDS: rows [3604 .. 3745], 142 rows x 100 f32 = 56.8KB
// (stay <= 64KB: CU-mode default makes larger static LDS a launch-time risk;
//  this also keeps ~5 blocks/WGP of LDS headroom for occupancy)
constexpr int LDS_BASE_ROW = 3604;
constexpr int LDS_ROWS     = 142;
constexpr int LDS_FLOATS   = LDS_ROWS * D_EMB;   // 14200
constexpr int LDS_V4       = LDS_FLOATS / 4;     // 3550 (exact)
} // namespace

__global__ void __launch_bounds__(BLOCK)
spline_encoding_kernel(const int* __restrict__ coords,
                       const float* __restrict__ w,
                       float* __restrict__ out)
{
    __shared__ __align__(16) float lds_w[LDS_FLOATS];

    const int tid = threadIdx.x;

    // ---- Stage coarse tables (scales 4,5) global -> LDS asynchronously ----
    {
        const float* gsrc = w + LDS_BASE_ROW * D_EMB;   // 16B-aligned offset
#if __has_builtin(__builtin_amdgcn_global_load_async_to_lds_b128)
        GLOBAL_AS v4i_t* gvec = (GLOBAL_AS v4i_t*)gsrc;
        LOCAL_AS  v4i_t* lvec = (LOCAL_AS  v4i_t*)lds_w;
        for (int i = tid; i < LDS_V4; i += BLOCK) {
            __builtin_amdgcn_global_load_async_to_lds_b128(
                gvec + i, lvec + i, /*offset=*/0, /*cpol=*/0);
        }
#if __has_builtin(__builtin_amdgcn_s_wait_asynccnt)
        __builtin_amdgcn_s_wait_asynccnt(0);
#else
        asm volatile("s_wait_asynccnt 0" ::: "memory");
#endif
#else
        for (int i = tid; i < LDS_V4; i += BLOCK) {
            float4 v = ((const float4*)gsrc)[i];
            ((float4*)lds_w)[i] = v;
        }
#endif
    }
    __syncthreads();

    // ---- Per-fragment gather + lerp: one fragment per wave32 ----
    const int wave = tid >> 5;           // which fragment of the 8-wide group
    const int lane = tid & 31;           // lanes 0..24 own float4 chunks of d
    const int base = blockIdx.x * FPB;

    // Software-pipelined coordinate fetch (coords -> idx -> gather is the
    // only serial chain; overlap next iteration's fetch with this compute).
    int c0 = coords[2 * (base + wave) + 0];
    int c1 = coords[2 * (base + wave) + 1];

    for (int j = 0; j < JITERS; ++j) {
        const int frag = base + j * WPB + wave;

        int nc0 = 0, nc1 = 0;
        if (j + 1 < JITERS) {
            nc0 = coords[2 * (frag + WPB) + 0];
            nc1 = coords[2 * (frag + WPB) + 1];
        }
        if (j + 2 < JITERS) {
            __builtin_prefetch(&coords[2 * (frag + 2 * WPB)], 0, 0);
        }

        v4f acc = {0.0f, 0.0f, 0.0f, 0.0f};
#pragma unroll
        for (int s = 0; s < 6; ++s) {
#pragma unroll
            for (int c = 0; c < 2; ++c) {
                int coord = (c == 0) ? c0 : c1;
                coord = (coord < -100000) ? -100000 : coord;
                coord = (coord >  99999)  ?  99999  : coord;

                const int bw = kBW[s];
                int q = coord / bw;              // trunc toward zero (magic mul)
                int r = coord - q * bw;
                if (r < 0) { q -= 1; r += bw; }  // Python floor-div semantics

                // idx is wave-uniform: force it scalar so gathers become
                // saddr-form global_load_b128 and address math rides the SALU.
                const int   sidx = __builtin_amdgcn_readfirstlane(q - kSHIFT[s] + kCUM[s]);
                const float a    = (float)r * kINV[s];

                v4f w0, w1;
                if (s >= 4) {
                    const v4f* lrow =
                        (const v4f*)(lds_w + (sidx - LDS_BASE_ROW) * D_EMB) + lane;
                    w0 = lrow[0];
                    w1 = lrow[D_EMB / 4];       // +100 floats = next row
                } else {
                    const v4f* grow = (const v4f*)(w + sidx * D_EMB) + lane;
                    w0 = grow[0];
                    w1 = grow[D_EMB / 4];
                }
                acc += (1.0f - a) * w0 + a * w1;
            }
        }

        if (lane < 25) {
            // Output is write-once / never re-read: stream it (TH=NT)
            __builtin_nontemporal_store(acc, (v4f*)(out + frag * D_EMB) + lane);
        }

        c0 = nc0;
        c1 = nc1;
    }
}

extern "C" void kernel_launch(void* const* d_in, const int* in_sizes, int n_in,
                              void* d_out, int out_size, void* d_ws, size_t ws_size,
                              hipStream_t stream)
{
    (void)in_sizes; (void)n_in; (void)out_size; (void)d_ws; (void)ws_size;

    const int*   coords = (const int*)d_in[0];   // (F, 2) int32
    const float* w      = (const float*)d_in[1]; // (3746, 100) f32
    float*       out    = (float*)d_out;         // (F, 100) f32

    dim3 grid(F_TOT / FPB);   // 1024 blocks
    dim3 block(BLOCK);        // 256 threads = 8 wave32
    spline_encoding_kernel<<<grid, block, 0, stream>>>(coords, w, out);
}